// GruopFeature_6811818131731
// MI455X (gfx1250) — compile-verified
//
#include <hip/hip_runtime.h>
#include <hip/hip_bf16.h>

typedef __attribute__((ext_vector_type(2))) float v2f;
typedef __attribute__((ext_vector_type(8))) float v8f;

#define NPTS   2048
#define NBATCH 8
#define NSAMP  32
#define INCH   128
#define OUTCH  64
#define NSUP   4          // SUPPORT
#define NCOLS  256        // SUPPORT*OUT_CH (columns 64..319 of weights)

// ---------------------------------------------------------------------------
// Kernel 1: KNN — one block per query point. Distances in LDS, 32 rounds of
// parallel argmin extraction (self masked to +inf, matching top_k(k+1)[1:]).
// ---------------------------------------------------------------------------
__global__ __launch_bounds__(256)
void knn_kernel(const float* __restrict__ verts, int* __restrict__ nbr) {
    const int gid = blockIdx.x;            // b*2048 + v
    const int b   = gid >> 11;
    const int vi  = gid & (NPTS - 1);
    const int tid = threadIdx.x;

    __shared__ float dist[NPTS];
    __shared__ float redv[256];
    __shared__ int   redi[256];

    const float* vb = verts + (size_t)b * NPTS * 3;
    const float px = vb[vi * 3 + 0];
    const float py = vb[vi * 3 + 1];
    const float pz = vb[vi * 3 + 2];

    for (int j = tid; j < NPTS; j += 256) {
        float dx = vb[j * 3 + 0] - px;
        float dy = vb[j * 3 + 1] - py;
        float dz = vb[j * 3 + 2] - pz;
        dist[j] = dx * dx + dy * dy + dz * dz;
    }
    __syncthreads();
    if (tid == 0) dist[vi] = 3.4e38f;      // exclude self
    __syncthreads();

    for (int s = 0; s < NSAMP; ++s) {
        float bestv = 3.4e38f; int besti = 0;
        for (int j = tid; j < NPTS; j += 256) {
            float d = dist[j];
            if (d < bestv) { bestv = d; besti = j; }
        }
        redv[tid] = bestv; redi[tid] = besti;
        __syncthreads();
        for (int off = 128; off > 0; off >>= 1) {
            if (tid < off) {
                if (redv[tid + off] < redv[tid]) {
                    redv[tid] = redv[tid + off];
                    redi[tid] = redi[tid + off];
                }
            }
            __syncthreads();
        }
        if (tid == 0) {
            int w = redi[0];
            nbr[(size_t)gid * NSAMP + s] = w;
            dist[w] = 3.4e38f;
        }
        __syncthreads();
    }
}

// ---------------------------------------------------------------------------
// Kernel 2: feature GEMM via V_WMMA_F32_16X16X4_F32.
// FS[16384][256] = feature_map[16384][128] @ weights[128][64..320] + bias
// One wave computes a 16x16 tile; 8 waves/block -> 16(M) x 128(N) strip.
// Grid: (16384/16, 256/128) = (1024, 2).
// ---------------------------------------------------------------------------
__global__ __launch_bounds__(256)
void gemm_wmma_kernel(const float* __restrict__ A, const float* __restrict__ W,
                      const float* __restrict__ bias, float* __restrict__ FS) {
    const int wave = threadIdx.x >> 5;                // 0..7
    const int lane = threadIdx.x & 31;
    const int row0 = blockIdx.x * 16;                 // M tile base
    const int col0 = blockIdx.y * 128 + wave * 16;    // N tile base in [0,256)

    // A fragment (16x4 f32): lane m = lane&15; VGPR0/1 = K=kk,kk+1 (kk = (lane>>4)*2)
    const int m  = lane & 15;
    const int kk = (lane >> 4) * 2;
    const float* arow = A + (size_t)(row0 + m) * INCH + kk;
    // B fragment (4x16 f32): lane n = lane&15; rows K=kk,kk+1
    const int n = lane & 15;
    const float* wcol = W + (size_t)kk * 320 + (OUTCH + col0 + n);

    v8f acc = {};
    #pragma unroll 4
    for (int k0 = 0; k0 < INCH; k0 += 4) {
        v2f a, bfr;
        a.x   = arow[k0];
        a.y   = arow[k0 + 1];
        bfr.x = wcol[(size_t)k0 * 320];
        bfr.y = wcol[(size_t)(k0 + 1) * 320];
        acc = __builtin_amdgcn_wmma_f32_16x16x4_f32(
                  /*neg_a=*/false, a, /*neg_b=*/false, bfr,
                  /*c_mod=*/(short)0, acc, /*reuse_a=*/false, /*reuse_b=*/false);
    }

    // D layout: VGPR j -> M=row0+j (lanes 0-15) / row0+j+8 (lanes 16-31), N=col0+(lane&15)
    const int dm = (lane < 16) ? 0 : 8;
    const int dn = lane & 15;
    const float bv = bias[OUTCH + col0 + dn];
    #pragma unroll
    for (int j = 0; j < 8; ++j) {
        FS[(size_t)(row0 + dm + j) * NCOLS + col0 + dn] = acc[j] + bv;
    }
}

// ---------------------------------------------------------------------------
// Kernel 3: q/k projections, softmax attention, support-mean, output.
// One block per (b,v); writes out[b][c][v][n] for c<64, n<32.
// ---------------------------------------------------------------------------
__global__ __launch_bounds__(256)
void attn_kernel(const float* __restrict__ verts,
                 const float* __restrict__ kvw, const float* __restrict__ kvb,
                 const float* __restrict__ qw,  const float* __restrict__ qb,
                 const int* __restrict__ nbr,   const float* __restrict__ FS,
                 float* __restrict__ out) {
    const int gid = blockIdx.x;
    const int b   = gid >> 11;
    const int vi  = gid & (NPTS - 1);
    const int tid = threadIdx.x;

    __shared__ float qs[OUTCH];
    __shared__ float part[256];
    __shared__ float theta[NSAMP];
    __shared__ float wgt[NSAMP];
    __shared__ int   nb[NSAMP];
    __shared__ float sval[2];

    const float* vb = verts + (size_t)b * NPTS * 3;
    const float px = vb[vi * 3 + 0];
    const float py = vb[vi * 3 + 1];
    const float pz = vb[vi * 3 + 2];

    if (tid < NSAMP) nb[tid] = nbr[(size_t)gid * NSAMP + tid];
    if (tid < OUTCH) {  // q projection: BC=64 channels, K=3
        qs[tid] = qw[tid * 3 + 0] * px + qw[tid * 3 + 1] * py +
                  qw[tid * 3 + 2] * pz + qb[tid];
    }
    __syncthreads();

    // theta[n] = (q . k_n)/sqrt(64); thread t handles neighbor t>>3, channels (t&7)*8..+8
    {
        const int n  = tid >> 3;
        const int c0 = (tid & 7) * 8;
        const int g  = nb[n];
        const float nx = vb[g * 3 + 0] - px;
        const float ny = vb[g * 3 + 1] - py;
        const float nz = vb[g * 3 + 2] - pz;
        float p = 0.f;
        #pragma unroll
        for (int c = c0; c < c0 + 8; ++c) {
            float kv = kvw[c * 3 + 0] * nx + kvw[c * 3 + 1] * ny +
                       kvw[c * 3 + 2] * nz + kvb[c];
            p += qs[c] * kv;
        }
        part[tid] = p;
    }
    __syncthreads();
    if (tid < NSAMP) {
        float s = 0.f;
        #pragma unroll
        for (int i = 0; i < 8; ++i) s += part[tid * 8 + i];
        theta[tid] = s * 0.125f;           // 1/sqrt(BC)=1/8
    }
    __syncthreads();
    if (tid == 0) {
        float mx = theta[0];
        for (int i = 1; i < NSAMP; ++i) mx = fmaxf(mx, theta[i]);
        sval[0] = mx;
    }
    __syncthreads();
    if (tid < NSAMP) theta[tid] = __expf(theta[tid] - sval[0]);
    __syncthreads();
    if (tid == 0) {
        float s = 0.f;
        for (int i = 0; i < NSAMP; ++i) s += theta[i];
        sval[1] = 1.f / s;
    }
    __syncthreads();
    if (tid < NSAMP) wgt[tid] = theta[tid] * sval[1];
    __syncthreads();

    // out[b][c][v][n] = wgt[n] * mean_s FS[b*2048+nb[n]][s*64+c]
    #pragma unroll
    for (int it = 0; it < 8; ++it) {
        const int o = it * 256 + tid;      // 0..2047
        const int c = o >> 5;              // 0..63
        const int n = o & 31;
        const float* fr = FS + ((size_t)b * NPTS + nb[n]) * NCOLS + c;
        const float s = fr[0] + fr[OUTCH] + fr[2 * OUTCH] + fr[3 * OUTCH];
        out[(((size_t)b * OUTCH + c) * NPTS + vi) * NSAMP + n] = wgt[n] * 0.25f * s;
    }
}

// ---------------------------------------------------------------------------
extern "C" void kernel_launch(void* const* d_in, const int* in_sizes, int n_in,
                              void* d_out, int out_size, void* d_ws, size_t ws_size,
                              hipStream_t stream) {
    (void)in_sizes; (void)n_in; (void)out_size; (void)ws_size;
    const float* verts   = (const float*)d_in[1];   // vertices (8,2048,3)
    const float* fmap    = (const float*)d_in[2];   // feature_map (8,2048,128)
    const float* weights = (const float*)d_in[3];   // (128, 320)
    const float* bias    = (const float*)d_in[4];   // (320,)
    const float* kvw     = (const float*)d_in[5];   // (128, 3) — only rows 0..63 used
    const float* kvb     = (const float*)d_in[6];   // (128,)
    const float* qw      = (const float*)d_in[7];   // (64, 3)
    const float* qb      = (const float*)d_in[8];   // (64,)
    float* out = (float*)d_out;

    const int rows = NBATCH * NPTS;                 // 16384
    float* FS  = (float*)d_ws;                                      // 16384*256 f32
    int*   nbr = (int*)((char*)d_ws + (size_t)rows * NCOLS * 4);    // 16384*32 i32

    knn_kernel<<<rows, 256, 0, stream>>>(verts, nbr);
    gemm_wmma_kernel<<<dim3(rows / 16, NCOLS / 128), 256, 0, stream>>>(fmap, weights, bias, FS);
    attn_kernel<<<rows, 256, 0, stream>>>(verts, kvw, kvb, qw, qb, nbr, FS, out);
}